// SupConLoss_46514495815719
// MI455X (gfx1250) — compile-verified
//
#include <hip/hip_runtime.h>
#include <hip/hip_bf16.h>
#include <math.h>

typedef float v2f __attribute__((ext_vector_type(2)));
typedef float v8f __attribute__((ext_vector_type(8)));

#define N_ROWS   8192
#define HALF_N   4096
#define DDIM     256
#define INV_TEMP 2.0f      // 1 / 0.5
#define SHIFT_C  2.0f      // softmax shift; = diag value, result invariant in it
#define EPS_F    1e-9f

// ---------------------------------------------------------------------------
// half-wave (16-lane) sum; xor offsets 1,2,4,8 stay inside each half
// ---------------------------------------------------------------------------
__device__ __forceinline__ float halfSum(float v) {
    #pragma unroll
    for (int o = 1; o < 16; o <<= 1) v += __shfl_xor(v, o, 32);
    return v;
}

// ---------------------------------------------------------------------------
// Kernel 1: row L2-normalization.  zn[r] = z[r] / max(||z[r]||, 1e-8)
// one wave per row, 8 rows per 256-thread block
// ---------------------------------------------------------------------------
__global__ __launch_bounds__(256) void supcon_normalize(
    const float* __restrict__ zi, const float* __restrict__ zj,
    float* __restrict__ zn)
{
    const int row  = blockIdx.x * 8 + (threadIdx.x >> 5);
    const int lane = threadIdx.x & 31;
    const float* src = (row < HALF_N) ? (zi + (size_t)row * DDIM)
                                      : (zj + (size_t)(row - HALF_N) * DDIM);
    float4 v0 = ((const float4*)src)[lane];
    float4 v1 = ((const float4*)src)[lane + 32];
    float ss = v0.x*v0.x + v0.y*v0.y + v0.z*v0.z + v0.w*v0.w
             + v1.x*v1.x + v1.y*v1.y + v1.z*v1.z + v1.w*v1.w;
    #pragma unroll
    for (int o = 1; o < 32; o <<= 1) ss += __shfl_xor(ss, o, 32);
    const float inv = 1.0f / fmaxf(sqrtf(ss), 1e-8f);
    float* dst = zn + (size_t)row * DDIM;
    v0.x *= inv; v0.y *= inv; v0.z *= inv; v0.w *= inv;
    v1.x *= inv; v1.y *= inv; v1.z *= inv; v1.w *= inv;
    ((float4*)dst)[lane]      = v0;
    ((float4*)dst)[lane + 32] = v1;
}

// ---------------------------------------------------------------------------
// Kernel 2: fused  sim = (zn zn^T)/T  +  masked softmax statistics.
// Block: 128 threads (4 waves), owns BM=64 rows. Wave w owns rows
// [I0+16w, I0+16w+16) and computes 4 column tiles (BN=64) per J step with
// V_WMMA_F32_16X16X4_F32.  Fixed softmax shift (c=2.0 >= all sims since rows
// are unit-norm) means every per-row statistic is a plain sum -> per-lane
// partials, cross-lane reduced ONCE after the J loop (no bpermute in loop).
// ---------------------------------------------------------------------------
#define BM   64
#define BN   64
#define STR  260   // 260 % 64 == 4 -> conflict-free b64 fragment loads

__global__ __launch_bounds__(128) void supcon_main(
    const float* __restrict__ zn, const int* __restrict__ labels,
    float* __restrict__ row_loss)
{
    __shared__ float shI[BM * STR];
    __shared__ float shJ[BN * STR];
    __shared__ int   shLab[BN];

    const int tid  = threadIdx.x;
    const int wave = tid >> 5;
    const int lane = tid & 31;
    const int half = lane >> 4;
    const int l15  = lane & 15;
    const int I0   = blockIdx.x * BM;

    // ---- stage the I panel once (reused for all 128 J iterations) ----
    for (int idx = tid; idx < BM * (DDIM / 4); idx += 128) {
        const int r = idx >> 6, c4 = idx & 63;
        float4 v = ((const float4*)(zn + (size_t)(I0 + r) * DDIM))[c4];
        *(float4*)&shI[r * STR + c4 * 4] = v;
    }

    // ---- per-lane partial state (rows p and p+8 on the two lane halves) ----
    float Sx[8], Ps[8], Cn[8];
    int   labR[8];
    #pragma unroll
    for (int p = 0; p < 8; ++p) {
        const int i = I0 + 16 * wave + 8 * half + p;
        labR[p] = labels[i & (HALF_N - 1)];
        Sx[p] = 0.f; Ps[p] = 0.f; Cn[p] = 0.f;
    }
    __syncthreads();

    for (int J0 = 0; J0 < N_ROWS; J0 += BN) {
        // ---- stage J panel + its labels ----
        for (int idx = tid; idx < BN * (DDIM / 4); idx += 128) {
            const int r = idx >> 6, c4 = idx & 63;
            float4 v = ((const float4*)(zn + (size_t)(J0 + r) * DDIM))[c4];
            *(float4*)&shJ[r * STR + c4 * 4] = v;
        }
        if (tid < BN) shLab[tid] = labels[(J0 + tid) & (HALF_N - 1)];
        // ---- prefetch next J panel into WGP cache (global_prefetch_b8) ----
        if (J0 + BN < N_ROWS) {
            const char* nxt = (const char*)(zn + (size_t)(J0 + BN) * DDIM);
            #pragma unroll
            for (int t = 0; t < 4; ++t)
                __builtin_prefetch(nxt + (size_t)(tid + t * 128) * 128, 0, 3);
        }
        __syncthreads();

        // ---- K loop: 64 steps of 16x16x4 fp32 WMMA, 4 column tiles ----
        v8f a0 = {}, a1 = {}, a2 = {}, a3 = {};
        const float* aBase = &shI[(16 * wave + l15) * STR + 2 * half];
        const float* bBase = &shJ[l15 * STR + 2 * half];
        #pragma unroll 4
        for (int k = 0; k < DDIM; k += 4) {
            v2f av = *(const v2f*)(aBase + k);
            v2f b0 = *(const v2f*)(bBase + k);
            v2f b1 = *(const v2f*)(bBase + 16 * STR + k);
            v2f b2 = *(const v2f*)(bBase + 32 * STR + k);
            v2f b3 = *(const v2f*)(bBase + 48 * STR + k);
            a0 = __builtin_amdgcn_wmma_f32_16x16x4_f32(false, av, false, b0, (short)0, a0, false, false);
            a1 = __builtin_amdgcn_wmma_f32_16x16x4_f32(false, av, false, b1, (short)0, a1, false, false);
            a2 = __builtin_amdgcn_wmma_f32_16x16x4_f32(false, av, false, b2, (short)0, a2, false, false);
            a3 = __builtin_amdgcn_wmma_f32_16x16x4_f32(false, av, false, b3, (short)0, a3, false, false);
        }

        // ---- per-lane accumulation (no cross-lane ops in the J loop) ----
        const int iBase = I0 + 16 * wave + 8 * half;
        const int lj0 = shLab[l15],      lj1 = shLab[16 + l15];
        const int lj2 = shLab[32 + l15], lj3 = shLab[48 + l15];
        const int j0 = J0 + l15,      j1 = J0 + 16 + l15;
        const int j2 = J0 + 32 + l15, j3 = J0 + 48 + l15;

        #pragma unroll
        for (int p = 0; p < 8; ++p) {
            const int i = iBase + p;
            const float s0 = INV_TEMP * a0[p], s1 = INV_TEMP * a1[p];
            const float s2 = INV_TEMP * a2[p], s3 = INV_TEMP * a3[p];

            const bool d0 = (i == j0), d1 = (i == j1);
            const bool d2 = (i == j2), d3 = (i == j3);
            const bool p0 = (labR[p] == lj0) && !d0;
            const bool p1 = (labR[p] == lj1) && !d1;
            const bool p2 = (labR[p] == lj2) && !d2;
            const bool p3 = (labR[p] == lj3) && !d3;

            Sx[p] += (d0 ? 0.f : __expf(s0 - SHIFT_C))
                   + (d1 ? 0.f : __expf(s1 - SHIFT_C))
                   + (d2 ? 0.f : __expf(s2 - SHIFT_C))
                   + (d3 ? 0.f : __expf(s3 - SHIFT_C));
            Ps[p] += (p0 ? s0 : 0.f) + (p1 ? s1 : 0.f)
                   + (p2 ? s2 : 0.f) + (p3 ? s3 : 0.f);
            Cn[p] += (p0 ? 1.f : 0.f) + (p1 ? 1.f : 0.f)
                   + (p2 ? 1.f : 0.f) + (p3 ? 1.f : 0.f);
        }
        __syncthreads();   // protect shJ before next iteration overwrites it
    }

    // ---- single cross-lane merge + per-row result ----
    #pragma unroll
    for (int p = 0; p < 8; ++p) {
        const float S = halfSum(Sx[p]);
        const float P = halfSum(Ps[p]);
        const float C = halfSum(Cn[p]);
        if (l15 == 0) {
            const int i = I0 + 16 * wave + 8 * half + p;
            // mean_log_prob_pos with shift c:  (P - C*c - C*log(S+eps)) / (C+eps)
            row_loss[i] = (P - C * (SHIFT_C + logf(S + EPS_F))) / (C + EPS_F);
        }
    }
}

// ---------------------------------------------------------------------------
// Kernel 3: deterministic reduction  out = -mean(row_loss)
// ---------------------------------------------------------------------------
__global__ __launch_bounds__(256) void supcon_finalize(
    const float* __restrict__ row_loss, float* __restrict__ out)
{
    __shared__ float sh[256];
    float s = 0.f;
    for (int i = threadIdx.x; i < N_ROWS; i += 256) s += row_loss[i];
    sh[threadIdx.x] = s;
    __syncthreads();
    for (int o = 128; o > 0; o >>= 1) {
        if (threadIdx.x < o) sh[threadIdx.x] += sh[threadIdx.x + o];
        __syncthreads();
    }
    if (threadIdx.x == 0) out[0] = -sh[0] / (float)N_ROWS;
}

// ---------------------------------------------------------------------------
extern "C" void kernel_launch(void* const* d_in, const int* in_sizes, int n_in,
                              void* d_out, int out_size, void* d_ws, size_t ws_size,
                              hipStream_t stream)
{
    const float* zi     = (const float*)d_in[0];
    const float* zj     = (const float*)d_in[1];
    const int*   labels = (const int*)d_in[2];
    float*       out    = (float*)d_out;

    float* zn       = (float*)d_ws;                    // 8192*256 floats = 8 MB
    float* row_loss = zn + (size_t)N_ROWS * DDIM;      // + 32 KB

    supcon_normalize<<<N_ROWS / 8, 256, 0, stream>>>(zi, zj, zn);
    supcon_main<<<N_ROWS / BM, 128, 0, stream>>>(zn, labels, row_loss);
    supcon_finalize<<<1, 256, 0, stream>>>(row_loss, out);
}